// HeteroSAGE_5815385719105
// MI455X (gfx1250) — compile-verified
//
#include <hip/hip_runtime.h>

#define NW 200000
#define NE 400000
#define D  256
#define DO 128

typedef float v2f __attribute__((ext_vector_type(2)));
typedef float v8f __attribute__((ext_vector_type(8)));

// ---------------------------------------------------------------- zero
__global__ void zero_f32(float* __restrict__ p, size_t n) {
  size_t i = (size_t)blockIdx.x * blockDim.x + threadIdx.x;
  size_t stride = (size_t)gridDim.x * blockDim.x;
  for (; i < n; i += stride) p[i] = 0.0f;
}

// ------------------------------------------------- edge scatter (mean num/den)
// One wave32 per edge: 32 lanes x 8 floats = 256-float row, coalesced atomics.
__global__ void scatter_add_mean(const float* __restrict__ X,
                                 const int* __restrict__ src,
                                 const int* __restrict__ dst,
                                 float* __restrict__ agg,
                                 float* __restrict__ cnt,
                                 int nEdges) {
  int wave = blockIdx.x * (blockDim.x >> 5) + (threadIdx.x >> 5);
  int lane = threadIdx.x & 31;
  if (wave >= nEdges) return;
  int s = src[wave];
  int d = dst[wave];
  const float* xs = X + (size_t)s * D;
  float* ad = agg + (size_t)d * D;
  __builtin_prefetch(xs + lane, 0, 1);   // global_prefetch_b8
  if (lane == 0) atomicAdd(cnt + d, 1.0f);
#pragma unroll
  for (int j = 0; j < D / 32; ++j)
    atomicAdd(ad + lane + 32 * j, xs[lane + 32 * j]);
}

// ------------------------------------------------- agg /= max(cnt,1)
__global__ void finalize_mean(float* __restrict__ agg,
                              const float* __restrict__ cnt, size_t n) {
  size_t i = (size_t)blockIdx.x * blockDim.x + threadIdx.x;
  size_t stride = (size_t)gridDim.x * blockDim.x;
  for (; i < n; i += stride) {
    float c = cnt[i >> 8];            // D == 256
    agg[i] = agg[i] / fmaxf(c, 1.0f);
  }
}

// ------------------------------------------------- fragment bundle per k-step
template <int DUAL, int NT>
struct Frags {
  v2f a, a2;
  v2f b1[NT];
  v2f b2[NT];
};

template <int DUAL, int NT>
__device__ __forceinline__ void load_frags(const float* __restrict__ A1,
                                           const float* __restrict__ A2,
                                           const float* __restrict__ W1,
                                           const float* __restrict__ W2,
                                           int aoff, int woff, int k,
                                           Frags<DUAL, NT>& f) {
  f.a = *(const v2f*)(A1 + aoff + k);
  if (DUAL) f.a2 = *(const v2f*)(A2 + aoff + k);
#pragma unroll
  for (int t = 0; t < NT; ++t) {
    f.b1[t] = *(const v2f*)(W1 + woff + t * 16 * D + k);
    if (DUAL) f.b2[t] = *(const v2f*)(W2 + woff + t * 16 * D + k);
  }
}

template <int DUAL, int NT>
__device__ __forceinline__ void mma_frags(const Frags<DUAL, NT>& f,
                                          v8f (&acc)[NT]) {
#pragma unroll
  for (int t = 0; t < NT; ++t) {
    acc[t] = __builtin_amdgcn_wmma_f32_16x16x4_f32(false, f.a, false, f.b1[t],
                                                   (short)0, acc[t],
                                                   false, false);
    if (DUAL)
      acc[t] = __builtin_amdgcn_wmma_f32_16x16x4_f32(false, f.a2, false,
                                                     f.b2[t], (short)0, acc[t],
                                                     false, false);
  }
}

// ------------------------------------------------- fused SAGE GEMM via WMMA f32
// out[m, n] = act( sum_k A1[m,k]*W1[n,k] (+ A2[m,k]*W2[n,k]) + bias[n] )
// Each wave owns a 16 x (16*NT) output strip; K loop of 4 via
// V_WMMA_F32_16X16X4_F32. Software-pipelined: fragments for k-step k+4 are
// loaded BEFORE the WMMAs of k-step k issue, so every WMMA waits on loads
// that have been in flight for a full iteration (s_wait_loadcnt ladders
// instead of round-tripping to 0 per load).
// A frag  (16x4):  lane L holds A[L%16][2*(L/16) + {0,1}]  -> one float2 load
// B frag  (4x16):  B[k][n] = W[n][k]; lane L holds W-row L%16, k pair 2*(L/16)
// C/D     (16x16): vgpr r, lane L -> element [r + 8*(L/16)][L%16]
template <int DUAL, int RELU, int NT>
__global__ void wmma_sage_gemm(const float* __restrict__ A1,
                               const float* __restrict__ W1,
                               const float* __restrict__ A2,
                               const float* __restrict__ W2,
                               const float* __restrict__ bias,
                               float* __restrict__ out,
                               int groupsN, int nCols) {
  int wave  = blockIdx.x * (blockDim.x >> 5) + (threadIdx.x >> 5);
  int lane  = threadIdx.x & 31;
  int tileM = wave / groupsN;
  int group = wave - tileM * groupsN;
  int tileN0 = group * NT;
  int half  = lane >> 4;      // selects K pair {0,1} vs {2,3}
  int lidx  = lane & 15;

  // 32-bit element offsets (max ~51M << 2^31) -> saddr + voffset addressing
  int aoff = (tileM * 16 + lidx) * D + 2 * half;
  int woff = (tileN0 * 16 + lidx) * D + 2 * half;

  v8f acc[NT];
#pragma unroll
  for (int t = 0; t < NT; ++t) acc[t] = {};

  Frags<DUAL, NT> cur, nxt;
  load_frags<DUAL, NT>(A1, A2, W1, W2, aoff, woff, 0, cur);

#pragma unroll 2
  for (int k = 0; k < D - 4; k += 4) {
    load_frags<DUAL, NT>(A1, A2, W1, W2, aoff, woff, k + 4, nxt);
    mma_frags<DUAL, NT>(cur, acc);
    cur = nxt;
  }
  mma_frags<DUAL, NT>(cur, acc);   // epilogue, no trailing prefetch (OOB-safe)

  int obase = (tileM * 16 + 8 * half) * nCols + lidx;
#pragma unroll
  for (int t = 0; t < NT; ++t) {
    float bv = bias[(tileN0 + t) * 16 + lidx];
    int ooff = obase + (tileN0 + t) * 16;
#pragma unroll
    for (int r = 0; r < 8; ++r) {
      float v = acc[t][r] + bv;
      if (RELU) v = fmaxf(v, 0.0f);
      out[ooff + r * nCols] = v;
    }
  }
}

// ---------------------------------------------------------------- launcher
extern "C" void kernel_launch(void* const* d_in, const int* in_sizes, int n_in,
                              void* d_out, int out_size, void* d_ws, size_t ws_size,
                              hipStream_t stream) {
  (void)in_sizes; (void)n_in; (void)out_size; (void)ws_size;

  const float* x_word = (const float*)d_in[0];
  const float* Wl     = (const float*)d_in[3];   // [2,3,256,256]
  const float* bl     = (const float*)d_in[4];   // [2,3,256]
  const float* Wr     = (const float*)d_in[5];   // [2,3,256,256]
  const float* fcW    = (const float*)d_in[6];   // [128,256]
  const float* fcb    = (const float*)d_in[7];   // [128]
  const int* dep_src  = (const int*)d_in[8];
  const int* dep_dst  = (const int*)d_in[9];
  float* out = (float*)d_out;

  // workspace: cnt | agg | h | g  (contiguous; cnt+agg zeroed together)
  float* cnt = (float*)d_ws;
  float* agg = cnt + NW;
  float* h   = agg + (size_t)NW * D;
  float* g   = h   + (size_t)NW * D;
  size_t aggN = (size_t)NW * D;

  const float* Wl00 = Wl;                const float* Wr00 = Wr;
  const float* bl00 = bl;
  const float* Wl10 = Wl + 3 * D * D;    const float* Wr10 = Wr + 3 * D * D;
  const float* bl10 = bl + 3 * D;

  constexpr int NT = 4;                          // 64-wide N strip per wave
  int scatterBlocks = NE / 8;                    // 8 waves/block, exact
  int groupsD  = D  / (16 * NT);                 // 4
  int groupsDO = DO / (16 * NT);                 // 2
  int gemmBlocks = (NW / 16) * groupsD  / 8;     // 6250, exact (EXEC all-1s)
  int outBlocks  = (NW / 16) * groupsDO / 8;     // 3125, exact

  // ---- layer 1: mean aggregation over dep edges
  zero_f32<<<2048, 256, 0, stream>>>(cnt, (size_t)NW + aggN);
  scatter_add_mean<<<scatterBlocks, 256, 0, stream>>>(x_word, dep_src, dep_dst,
                                                      agg, cnt, NE);
  finalize_mean<<<4096, 256, 0, stream>>>(agg, cnt, aggN);
  // h = relu(mean1 @ Wl00^T + x_word @ Wr00^T + bl00)
  wmma_sage_gemm<1, 1, NT><<<gemmBlocks, 256, 0, stream>>>(
      agg, Wl00, x_word, Wr00, bl00, h, groupsD, D);

  // ---- layer 2: same edge structure, source features = h
  zero_f32<<<2048, 256, 0, stream>>>(cnt, (size_t)NW + aggN);
  scatter_add_mean<<<scatterBlocks, 256, 0, stream>>>(h, dep_src, dep_dst,
                                                      agg, cnt, NE);
  finalize_mean<<<4096, 256, 0, stream>>>(agg, cnt, aggN);
  // g = relu(mean2 @ Wl10^T + h @ Wr10^T + bl10)
  wmma_sage_gemm<1, 1, NT><<<gemmBlocks, 256, 0, stream>>>(
      agg, Wl10, h, Wr10, bl10, g, groupsD, D);

  // ---- fc head: out = g @ fcW^T + fcb
  wmma_sage_gemm<0, 0, NT><<<outBlocks, 256, 0, stream>>>(
      g, fcW, nullptr, nullptr, fcb, out, groupsDO, DO);
}